// RBFKernel_25331717112421
// MI455X (gfx1250) — compile-verified
//
#include <hip/hip_runtime.h>

// RBF kernel: out[b,n,m] = exp(-max(||z_m - mu_bn||^2, 0) / (2*clamp(sigma)^2))
// Fixed shapes: z [4096,2], mu [16384,2], sigma [16384], out [16384,4096] fp32.
//
// Store-bandwidth bound (256 MB @ 23.3 TB/s ~ 11us). Distance tiles via
// V_WMMA_F32_16X16X4_F32 with the augmented-vector trick:
//   A row = [-2*mux, -2*muy, |mu|^2, 1],  B col = [zx, zy, 1, |z|^2]^T
//   => D[r][c] = |mu_r - z_c|^2 exactly, one WMMA per 16x16 tile.
// M is compile-time so all 128 store addresses per wave fold into the 24-bit
// immediate offset of a single base address (no 64-bit VALU address chains).

typedef float v2f __attribute__((ext_vector_type(2)));
typedef float v8f __attribute__((ext_vector_type(8)));

#if __has_builtin(__builtin_amdgcn_exp2f)
#define EXP2F(x) __builtin_amdgcn_exp2f(x)   // v_exp_f32
#else
#define EXP2F(x) exp2f(x)
#endif

#if __has_builtin(__builtin_amdgcn_rcpf)
#define RCPF(x) __builtin_amdgcn_rcpf(x)     // v_rcp_f32
#else
#define RCPF(x) (1.0f / (x))
#endif

#if __has_builtin(__builtin_amdgcn_fmed3f)
#define CLAMP0(x) __builtin_amdgcn_fmed3f((x), 0.0f, 3.402823466e38f) // 1x v_med3
#else
#define CLAMP0(x) fmaxf((x), 0.0f)
#endif

#define MDIM    4096                         // grid points (z rows)
#define MIN_SIG 0.1f
#define MAX_SIG 10.0f
#define NLOG2E  (-1.44269504088896340736f)   // -log2(e)

__global__ __launch_bounds__(256) void rbf_wmma_kernel(
    const float* __restrict__ z,      // [MDIM,2]
    const float* __restrict__ mu,     // [BN,2]
    const float* __restrict__ sigma,  // [BN]
    float* __restrict__ out)          // [BN,MDIM]
{
  const int lane   = threadIdx.x & 31;
  const int col    = lane & 15;
  const int hiHalf = lane >> 4;                        // 0 or 1

  const int w        = (blockIdx.x * 256 + (int)threadIdx.x) >> 5;
  const int rowTile  = w >> 4;                         // 16-token tile
  const int colChunk = w & 15;                         // 256 m-values
  const int n_base   = rowTile << 4;
  const int m_start  = colChunk << 8;

  // ---------- A operand (held for entire wave lifetime) ----------
  // 16x4 fp32 A: lanes 0-15 hold K=0,1 ; lanes 16-31 hold K=2,3 (row = lane%16)
  const int tok = n_base + col;
  const float2 mv = *reinterpret_cast<const float2*>(mu + 2 * tok);
  const float musq = mv.x * mv.x + mv.y * mv.y;
  v2f a;
  a.x = hiHalf ? musq : (-2.0f * mv.x);                // K=2 : K=0
  a.y = hiHalf ? 1.0f : (-2.0f * mv.y);                // K=3 : K=1

  // ---------- per-lane exp coefficients for the 8 D rows it owns ----------
  // D VGPR k: lanes 0-15 -> row k, lanes 16-31 -> row k+8
  const int rbase = n_base + (hiHalf << 3);
  const float4* sp = reinterpret_cast<const float4*>(sigma + rbase);
  const float4 s0 = sp[0];
  const float4 s1 = sp[1];
  float coef[8];
  {
    float sv[8] = {s0.x, s0.y, s0.z, s0.w, s1.x, s1.y, s1.z, s1.w};
#pragma unroll
    for (int k = 0; k < 8; ++k) {
      float s = fminf(fmaxf(sv[k], MIN_SIG), MAX_SIG);  // -> v_med3_num_f32
      coef[k] = NLOG2E * RCPF(2.0f * s * s);            // mul + v_rcp + mul
    }
  }

  // Single base addresses; all per-tile/per-row deltas become immediate offsets
  // (max byte delta = 7*16384 + 15*64 = 115648 < 2^23).
  const float* zbase = z + 2 * (m_start + col);
  float* obase = out + (size_t)rbase * MDIM + (m_start + col);

  // ---------- sweep 16 column tiles (B operand is tiny, L0-resident) ----------
#pragma unroll
  for (int t = 0; t < 16; ++t) {
    const float2 zv = *reinterpret_cast<const float2*>(zbase + 32 * t);
    v2f b;
    b.x = hiHalf ? 1.0f : zv.x;                        // K=2 : K=0
    b.y = hiHalf ? (zv.x * zv.x + zv.y * zv.y) : zv.y; // K=3 : K=1

    v8f c = {};
    // D = A x B : full 16x16 tile of squared distances in one instruction
    v8f d = __builtin_amdgcn_wmma_f32_16x16x4_f32(
        /*neg_a=*/false, a, /*neg_b=*/false, b,
        /*c_mod=*/(short)0, c, /*reuse_a=*/false, /*reuse_b=*/false);

#pragma unroll
    for (int k = 0; k < 8; ++k) {
      const float d2 = CLAMP0(d[k]);                   // one v_med3_num_f32
      const float v  = EXP2F(d2 * coef[k]);            // v_mul + v_exp_f32
      // 256 MB write-once output > 192 MB L2: bypass with non-temporal store.
      // Constant index -> folds into the store's immediate offset field.
      __builtin_nontemporal_store(v, obase + k * MDIM + t * 16);
    }
  }
}

extern "C" void kernel_launch(void* const* d_in, const int* in_sizes, int n_in,
                              void* d_out, int out_size, void* d_ws, size_t ws_size,
                              hipStream_t stream) {
  const float* z     = (const float*)d_in[0];   // [M,2]
  const float* mu    = (const float*)d_in[1];   // [B,N,2]
  const float* sigma = (const float*)d_in[2];   // [B,N,1]
  float* out = (float*)d_out;

  const int BN = in_sizes[2];                   // 16384 tokens
  const int rowTiles  = BN / 16;                // 1024
  const int colChunks = MDIM / 256;             // 16
  const int totalWaves = rowTiles * colChunks;  // 16384
  const int threads = 256;                      // 8 waves/block
  const int blocks  = totalWaves / (threads / 32);  // 2048

  rbf_wmma_kernel<<<blocks, threads, 0, stream>>>(z, mu, sigma, out);
}